// MGNN_23656679866486
// MI455X (gfx1250) — compile-verified
//
#include <hip/hip_runtime.h>
#include <hip/hip_bf16.h>
#include <math.h>

typedef __attribute__((ext_vector_type(2))) float v2f;
typedef __attribute__((ext_vector_type(8))) float v8f;

#define D_IN   128
#define HID    256
#define LG_DIN 7
#define LG_HID 8

// ---------------------------------------------------------------------------
// degree count: one thread per edge, f32 atomic into cnt[dst]
// ---------------------------------------------------------------------------
__global__ __launch_bounds__(256)
void count_kernel(const int* __restrict__ dst, int E, float* __restrict__ cnt) {
    int i = blockIdx.x * blockDim.x + threadIdx.x;
    if (i < E) atomicAdd(&cnt[dst[i]], 1.0f);
}

// ---------------------------------------------------------------------------
// scatter-sum: one thread per (edge, float4 chunk); feat rows are D = 1<<dshift
// ---------------------------------------------------------------------------
__global__ __launch_bounds__(256)
void scatter_sum_kernel(const float* __restrict__ feat,
                        const int* __restrict__ src, const int* __restrict__ dst,
                        int E, int dshift, float* __restrict__ out) {
    const int cshift = dshift - 2;                       // chunks of 4 floats
    long long idx = (long long)blockIdx.x * blockDim.x + threadIdx.x;
    int e = (int)(idx >> cshift);
    if (e >= E) return;
    int c = (int)(idx & ((1 << cshift) - 1));
    const float4 v = *(const float4*)(feat + ((size_t)src[e] << dshift) + c * 4);
    float* o = out + ((size_t)dst[e] << dshift) + c * 4;
    atomicAdd(o + 0, v.x);
    atomicAdd(o + 1, v.y);
    atomicAdd(o + 2, v.z);
    atomicAdd(o + 3, v.w);
}

// ---------------------------------------------------------------------------
// mean scale: x[row][*] /= max(cnt[row], 1)
// ---------------------------------------------------------------------------
__global__ __launch_bounds__(256)
void mean_scale_kernel(float* __restrict__ x, const float* __restrict__ cnt,
                       int rows, int dshift) {
    long long i = (long long)blockIdx.x * blockDim.x + threadIdx.x;
    if (i >= ((long long)rows << dshift)) return;
    float c = cnt[i >> dshift];
    x[i] = x[i] / fmaxf(c, 1.0f);
}

// ---------------------------------------------------------------------------
// deterministic dropout hash
// ---------------------------------------------------------------------------
__device__ __forceinline__ unsigned mix_hash(unsigned a, unsigned b, unsigned c) {
    unsigned h = a * 0x9E3779B1u ^ b * 0x85EBCA77u ^ c * 0xC2B2AE3Du;
    h ^= h >> 16; h *= 0x7FEB352Du;
    h ^= h >> 15; h *= 0x846CA68Bu;
    h ^= h >> 16;
    return h;
}

// ---------------------------------------------------------------------------
// Multi-term WMMA GEMM:  C[M x Nout] = sum_i A_i[M x K] @ B_i[K x Nout] (+bias)
//
// Block = 128 threads = 4 wave32. Block tile: 16 rows x 64 cols; each wave
// owns one 16x16 C tile (V_WMMA_F32_16X16X4_F32, exact fp32).
// Per K-chunk (BK=64) the block stages into LDS:
//   sA[t][16][68]        A tile, padded row stride 68 -> conflict-free frags
//   sB[t][BK/2][64][2]   B tile as interleaved (k, k+1) pairs -> one aligned
//                        ds_load_b64 per fragment
// Inner loop is divergence-free (EXEC all ones for WMMA); N-boundary guard is
// applied at LDS fill (pad cols = 0) and at the C store.
// epilogue: 0 = none, 1 = relu + dropout(p=0.5, x2 scale, hash keyed by seed)
// ---------------------------------------------------------------------------
template <int NTERMS>
__global__ __launch_bounds__(128)
void gemm_wmma_kernel(const float* __restrict__ A0, const float* __restrict__ B0,
                      const float* __restrict__ A1, const float* __restrict__ B1,
                      const float* __restrict__ A2, const float* __restrict__ B2,
                      const float* __restrict__ A3, const float* __restrict__ B3,
                      const float* __restrict__ bias0, const float* __restrict__ bias1,
                      float* __restrict__ C, int M, int K, int Nout,
                      int epilogue, unsigned seed) {
    constexpr int BK  = 64;
    constexpr int APAD = 68;                         // 16 rows x 68 (pad vs 64-bank LDS)
    __shared__ __align__(16) float sA[NTERMS * 16 * APAD];
    __shared__ __align__(16) float sB[NTERMS * BK * 64];

    const float* As[4] = {A0, A1, A2, A3};
    const float* Bs[4] = {B0, B1, B2, B3};

    const int lane = threadIdx.x & 31;
    const int wave = threadIdx.x >> 5;
    const int row0 = blockIdx.x * 16;
    const int colb = blockIdx.y * 64;                // block's first column
    const int m    = lane & 15;                      // A frag: M index
    const int koff = (lane >> 4) << 1;               // A/B frag: K pair select (0 or 2)
    const int nl   = wave * 16 + (lane & 15);        // local column within 64
    const int gn   = colb + nl;                      // global column

    v8f acc = {0.f, 0.f, 0.f, 0.f, 0.f, 0.f, 0.f, 0.f};

    for (int kb = 0; kb < K; kb += BK) {
        if (kb) __syncthreads();

        // ---- stage A tiles (float4 global loads) ----
        for (int idx = threadIdx.x; idx < NTERMS * 16 * (BK / 4); idx += 128) {
            const int t   = idx / (16 * (BK / 4));
            const int rem = idx - t * (16 * (BK / 4));
            const int mr  = rem / (BK / 4);
            const int k4  = (rem - mr * (BK / 4)) * 4;
            const float4 v = *(const float4*)(As[t] + (size_t)(row0 + mr) * K + kb + k4);
            float* d = &sA[t * 16 * APAD + mr * APAD + k4];
            d[0] = v.x; d[1] = v.y; d[2] = v.z; d[3] = v.w;
        }
        // ---- stage B tiles (coalesced rows, interleave k-pairs, zero-pad N) ----
        for (int idx = threadIdx.x; idx < NTERMS * BK * 64; idx += 128) {
            const int t   = idx / (BK * 64);
            const int rem = idx - t * (BK * 64);
            const int k   = rem >> 6;
            const int n   = rem & 63;
            const int gc  = colb + n;
            const float v = (gc < Nout) ? Bs[t][(size_t)(kb + k) * Nout + gc] : 0.f;
            sB[t * BK * 64 + (k >> 1) * 128 + n * 2 + (k & 1)] = v;
        }
        __syncthreads();

        // ---- divergence-free WMMA inner loop, all operands from LDS ----
#pragma unroll
        for (int k0 = 0; k0 < BK; k0 += 4) {
            const int ka = k0 + koff;
#pragma unroll
            for (int t = 0; t < NTERMS; ++t) {
                v2f a = *(const v2f*)&sA[t * 16 * APAD + m * APAD + ka];
                v2f b = *(const v2f*)&sB[t * BK * 64 + (ka >> 1) * 128 + nl * 2];
                acc = __builtin_amdgcn_wmma_f32_16x16x4_f32(false, a, false, b,
                                                            (short)0, acc, false, false);
            }
        }
    }

    if (gn >= Nout) return;
    float badd = 0.f;
    if (bias0) badd += bias0[gn];
    if (bias1) badd += bias1[gn];

    const int rbase = row0 + ((lane >> 4) << 3);     // C: VGPR r -> M=r / M=r+8
#pragma unroll
    for (int r = 0; r < 8; ++r) {
        const int row = rbase + r;
        float v = acc[r] + badd;
        if (epilogue == 1) {
            v = v > 0.f ? v : 0.f;
            unsigned h = mix_hash(seed, (unsigned)row, (unsigned)gn);
            v = (h & 1u) ? v * 2.0f : 0.0f;
        }
        C[(size_t)row * Nout + gn] = v;
    }
}

// ---------------------------------------------------------------------------
// log-softmax, one wave32 per row (ncol <= 256)
// ---------------------------------------------------------------------------
__global__ __launch_bounds__(128)
void log_softmax_kernel(float* __restrict__ x, int rows, int ncol) {
    const int lane = threadIdx.x & 31;
    const int row  = blockIdx.x * 4 + (threadIdx.x >> 5);
    if (row >= rows) return;
    float* p = x + (size_t)row * ncol;

    float mx = -INFINITY;
    for (int c = lane; c < ncol; c += 32) mx = fmaxf(mx, p[c]);
    for (int o = 16; o > 0; o >>= 1) mx = fmaxf(mx, __shfl_xor(mx, o, 32));

    float s = 0.f;
    for (int c = lane; c < ncol; c += 32) s += expf(p[c] - mx);
    for (int o = 16; o > 0; o >>= 1) s += __shfl_xor(s, o, 32);

    float l = logf(s);
    for (int c = lane; c < ncol; c += 32) p[c] = p[c] - mx - l;
}

// ---------------------------------------------------------------------------
// host orchestration
// ---------------------------------------------------------------------------
static inline unsigned ceil_div_u(long long a, long long b) {
    return (unsigned)((a + b - 1) / b);
}

extern "C" void kernel_launch(void* const* d_in, const int* in_sizes, int n_in,
                              void* d_out, int out_size, void* d_ws, size_t ws_size,
                              hipStream_t stream) {
    (void)n_in; (void)out_size; (void)ws_size;

    const float* xa   = (const float*)d_in[0];
    const float* xt   = (const float*)d_in[1];
    const int*   e_at = (const int*)d_in[2];
    const int*   e_ta = (const int*)d_in[3];
    const int*   e_aa = (const int*)d_in[4];
    const float* Wl1  = (const float*)d_in[5];   // [3,3,128,256]
    const float* Wr1  = (const float*)d_in[6];   // [3,3,128,256]
    const float* b1   = (const float*)d_in[7];   // [3,3,256]
    const float* Wl2h[3] = {(const float*)d_in[8],  (const float*)d_in[11], (const float*)d_in[14]};
    const float* Wr2h[3] = {(const float*)d_in[9],  (const float*)d_in[12], (const float*)d_in[15]};
    const float* b2h[3]  = {(const float*)d_in[10], (const float*)d_in[13], (const float*)d_in[16]};

    const int NA  = in_sizes[0] / D_IN;          // 120000
    const int NT  = in_sizes[1] / D_IN;          // 30000
    const int Eat = in_sizes[2] / 2;
    const int Eta = in_sizes[3] / 2;
    const int Eaa = in_sizes[4] / 2;
    const int dims[3] = {in_sizes[10] / 3, in_sizes[13] / 3, in_sizes[16] / 3}; // 129/32/18

    float* ws = (float*)d_ws;
    size_t off = 0;
    float* cnt_at = ws + off; off += (size_t)NT;                 // tag in-degree (at)
    float* cnt_ta = ws + off; off += (size_t)NA;                 // artwork in-degree (ta)
    float* cnt_aa = ws + off; off += (size_t)NA;                 // artwork in-degree (aa)
    float* m_at   = ws + off; off += (size_t)NT * D_IN;          // L1 means (shared over g)
    float* m_ta   = ws + off; off += (size_t)NA * D_IN;
    float* m_aa   = ws + off; off += (size_t)NA * D_IN;
    const size_t zero1_floats = off;                             // counts + L1 sums
    float* h_t    = ws + off; off += (size_t)NT * HID;           // per-g hidden (reused)
    float* h_a    = ws + off; off += (size_t)NA * HID;
    const size_t zero2_base = off;
    float* m2_at  = ws + off; off += (size_t)NT * HID;           // per-g L2 sums (reused)
    float* m2_ta  = ws + off; off += (size_t)NA * HID;
    float* m2_aa  = ws + off; off += (size_t)NA * HID;
    const size_t zero2_floats = off - zero2_base;

    const int BS = 256;

    // ---- counts + L1 segment sums (shared across the 3 GNNs) ----
    hipMemsetAsync(ws, 0, zero1_floats * sizeof(float), stream);

    count_kernel<<<ceil_div_u(Eat, BS), BS, 0, stream>>>(e_at + Eat, Eat, cnt_at);
    count_kernel<<<ceil_div_u(Eta, BS), BS, 0, stream>>>(e_ta + Eta, Eta, cnt_ta);
    count_kernel<<<ceil_div_u(Eaa, BS), BS, 0, stream>>>(e_aa + Eaa, Eaa, cnt_aa);

    scatter_sum_kernel<<<ceil_div_u((long long)Eat << (LG_DIN - 2), BS), BS, 0, stream>>>(
        xa, e_at, e_at + Eat, Eat, LG_DIN, m_at);
    scatter_sum_kernel<<<ceil_div_u((long long)Eta << (LG_DIN - 2), BS), BS, 0, stream>>>(
        xt, e_ta, e_ta + Eta, Eta, LG_DIN, m_ta);
    scatter_sum_kernel<<<ceil_div_u((long long)Eaa << (LG_DIN - 2), BS), BS, 0, stream>>>(
        xa, e_aa, e_aa + Eaa, Eaa, LG_DIN, m_aa);

    mean_scale_kernel<<<ceil_div_u((long long)NT << LG_DIN, BS), BS, 0, stream>>>(m_at, cnt_at, NT, LG_DIN);
    mean_scale_kernel<<<ceil_div_u((long long)NA << LG_DIN, BS), BS, 0, stream>>>(m_ta, cnt_ta, NA, LG_DIN);
    mean_scale_kernel<<<ceil_div_u((long long)NA << LG_DIN, BS), BS, 0, stream>>>(m_aa, cnt_aa, NA, LG_DIN);

    size_t out_off = 0;
    for (int g = 0; g < 3; ++g) {
        const int No = dims[g];
        const float* Wl1g = Wl1 + (size_t)g * 3 * D_IN * HID;    // + rel*D_IN*HID
        const float* Wr1g = Wr1 + (size_t)g * 3 * D_IN * HID;
        const float* b1g  = b1  + (size_t)g * 3 * HID;           // + rel*HID
        const float* Wl2  = Wl2h[g];                             // [3, HID, No]
        const float* Wr2  = Wr2h[g];
        const float* b2   = b2h[g];                              // [3, No]

        // ---- layer 1: h_t = relu+drop(m_at@Wl1[0] + xt@Wr1[0] + b1[0]) ----
        {
            dim3 grid(NT / 16, ceil_div_u(HID, 64));
            gemm_wmma_kernel<2><<<grid, 128, 0, stream>>>(
                m_at, Wl1g + 0 * (size_t)D_IN * HID,
                xt,   Wr1g + 0 * (size_t)D_IN * HID,
                nullptr, nullptr, nullptr, nullptr,
                b1g + 0 * HID, nullptr,
                h_t, NT, D_IN, HID, /*epilogue=*/1, /*seed=*/2000u + g);
        }
        // ---- layer 1: h_a = relu+drop(m_ta@Wl1[1] + xa@Wr1[1] + b1[1]
        //                              + m_aa@Wl1[2] + xa@Wr1[2] + b1[2]) ----
        {
            dim3 grid(NA / 16, ceil_div_u(HID, 64));
            gemm_wmma_kernel<4><<<grid, 128, 0, stream>>>(
                m_ta, Wl1g + 1 * (size_t)D_IN * HID,
                xa,   Wr1g + 1 * (size_t)D_IN * HID,
                m_aa, Wl1g + 2 * (size_t)D_IN * HID,
                xa,   Wr1g + 2 * (size_t)D_IN * HID,
                b1g + 1 * HID, b1g + 2 * HID,
                h_a, NA, D_IN, HID, /*epilogue=*/1, /*seed=*/1000u + g);
        }

        // ---- layer 2 segment means on hidden states ----
        hipMemsetAsync(ws + zero2_base, 0, zero2_floats * sizeof(float), stream);

        scatter_sum_kernel<<<ceil_div_u((long long)Eat << (LG_HID - 2), BS), BS, 0, stream>>>(
            h_a, e_at, e_at + Eat, Eat, LG_HID, m2_at);
        scatter_sum_kernel<<<ceil_div_u((long long)Eta << (LG_HID - 2), BS), BS, 0, stream>>>(
            h_t, e_ta, e_ta + Eta, Eta, LG_HID, m2_ta);
        scatter_sum_kernel<<<ceil_div_u((long long)Eaa << (LG_HID - 2), BS), BS, 0, stream>>>(
            h_a, e_aa, e_aa + Eaa, Eaa, LG_HID, m2_aa);

        mean_scale_kernel<<<ceil_div_u((long long)NT << LG_HID, BS), BS, 0, stream>>>(m2_at, cnt_at, NT, LG_HID);
        mean_scale_kernel<<<ceil_div_u((long long)NA << LG_HID, BS), BS, 0, stream>>>(m2_ta, cnt_ta, NA, LG_HID);
        mean_scale_kernel<<<ceil_div_u((long long)NA << LG_HID, BS), BS, 0, stream>>>(m2_aa, cnt_aa, NA, LG_HID);

        // ---- layer 2 logits straight into d_out, then in-place log-softmax ----
        float* o_a = (float*)d_out + out_off;            out_off += (size_t)NA * No;
        float* o_t = (float*)d_out + out_off;            out_off += (size_t)NT * No;

        {
            dim3 grid(NA / 16, ceil_div_u(No, 64));
            gemm_wmma_kernel<4><<<grid, 128, 0, stream>>>(
                m2_ta, Wl2 + 1 * (size_t)HID * No,
                h_a,   Wr2 + 1 * (size_t)HID * No,
                m2_aa, Wl2 + 2 * (size_t)HID * No,
                h_a,   Wr2 + 2 * (size_t)HID * No,
                b2 + 1 * No, b2 + 2 * No,
                o_a, NA, HID, No, /*epilogue=*/0, 0u);
        }
        {
            dim3 grid(NT / 16, ceil_div_u(No, 64));
            gemm_wmma_kernel<2><<<grid, 128, 0, stream>>>(
                m2_at, Wl2 + 0 * (size_t)HID * No,
                h_t,   Wr2 + 0 * (size_t)HID * No,
                nullptr, nullptr, nullptr, nullptr,
                b2 + 0 * No, nullptr,
                o_t, NT, HID, No, /*epilogue=*/0, 0u);
        }

        log_softmax_kernel<<<ceil_div_u(NA, 4), 128, 0, stream>>>(o_a, NA, No);
        log_softmax_kernel<<<ceil_div_u(NT, 4), 128, 0, stream>>>(o_t, NT, No);
    }
}